// MLMPS_28973849379436
// MI455X (gfx1250) — compile-verified
//
#include <hip/hip_runtime.h>
#include <math.h>

// ---------------------------------------------------------------------------
// MI455X (gfx1250, wave32) implementation of the 3-layer MPS classifier.
//
// Key algebra:
//   * layer-0/1 MPO norms cancel against the next layer's per-site input
//     normalization -> only norm2 is computed.
//   * layer-0 embedding norm is exactly sqrt(27)  (cos^2+sin^2 == 1).
//   * out0 = M0 . (T1 T2 ... T322 . last): T's are built batched with
//     v_wmma_f32_16x16x4_f32 (Amid_0 streamed from HBM exactly once),
//     reduced in chunks of 7 via 64x64x64 f32 WMMA products, then collapsed
//     onto a vector.
// ---------------------------------------------------------------------------

typedef float v2f __attribute__((ext_vector_type(2)));
typedef float v8f __attribute__((ext_vector_type(8)));

static __device__ __forceinline__ v8f wmma4(v2f a, v2f b, v8f c) {
  // D(16x16,f32) = A(16x4,f32) * B(4x16,f32) + C
  return __builtin_amdgcn_wmma_f32_16x16x4_f32(false, a, false, b,
                                               (short)0, c, false, false);
}

#define NB      64      // batch
#define NSITE0  324     // layer-0 sites
#define P0      54      // layer-0 phys dim
#define P0P     56      // padded (K multiple of 4)
#define NMID0   322     // layer-0 mid sites
#define DD      64      // bond dim
#define NCHUNK  46      // 46*7 == 322
#define CHL     7

// workspace layout (floats)
#define SZ_VT   ((size_t)NSITE0 * P0P * NB)          // v0^T: [site][p][b]
#define SZ_T    ((size_t)NMID0 * NB * DD * DD)       // T:    [site][b][d][e]
#define SZ_C    ((size_t)NCHUNK * NB * DD * DD)      // C:    [chunk][b][d][e]
#define SZ_H0   ((size_t)NB * NSITE0)
#define OFF_VT  ((size_t)0)
#define OFF_T   (OFF_VT + SZ_VT)
#define OFF_C   (OFF_T + SZ_T)
#define OFF_H0  (OFF_C + SZ_C)
#define OFF_NRM (OFF_H0 + SZ_H0)

// ---------------------------------------------------------------------------
// K1: trig embedding -> v0^T  [site][p(56, zero padded)][b]
// ---------------------------------------------------------------------------
__global__ void __launch_bounds__(256)
k_embed(const float* __restrict__ x, float* __restrict__ vT) {
  int tid = blockIdx.x * 256 + threadIdx.x;          // b fastest
  if (tid >= NB * NSITE0 * 27) return;
  int b = tid & 63;
  int rest = tid >> 6;
  int idx = rest % 27;
  int site = rest / 27;
  int si = site / 18, sj = site % 18;
  int ch = idx % 3, ac = idx / 3, cc = ac % 3, aa = ac / 3;
  float xv = x[(((size_t)b * 54 + (3 * si + aa)) * 54 + (3 * sj + cc)) * 3 + ch];
  float ang = 1.57079632679489662f * xv;
  const float inv = 0.19245008972987526f;            // 1/sqrt(27)
  vT[((size_t)site * P0P + idx) * NB + b]        = cosf(ang) * inv;
  vT[((size_t)site * P0P + 27 + idx) * NB + b]   = sinf(ang) * inv;
  if (idx < 2) vT[((size_t)site * P0P + 54 + idx) * NB + b] = 0.f;
}

// ---------------------------------------------------------------------------
// K2: T[n][b][d][e] = sum_p Amid0[n][d][e][p] * v[b][n+1][p]
//     as GEMM (4096 x 54pad56) @ (56 x 64 batches) with f32 WMMA.
//     grid: (4 de-quarters, 322 sites); Amid_0 streamed from HBM once.
// ---------------------------------------------------------------------------
__global__ void __launch_bounds__(256)
k_tmat(const float* __restrict__ Amid, const float* __restrict__ vT,
       float* __restrict__ T) {
  __shared__ float vs[P0P * NB];                     // 14 KB
  int site = blockIdx.y;                             // 0..321
  int q    = blockIdx.x;                             // 0..3
  int tid  = threadIdx.x;

  const float* vsrc = vT + (size_t)(site + 1) * P0P * NB;
  for (int i4 = tid; i4 < (P0P * NB) / 4; i4 += 256)
    ((float4*)vs)[i4] = ((const float4*)vsrc)[i4];
  __syncthreads();

  int wave = tid >> 5, lane = tid & 31;
  int m = lane & 15, h = lane >> 4;
  const float* Ab = Amid + (size_t)site * 4096 * P0;
  float* Tb = T + (size_t)site * NB * 4096;

  for (int rt8 = 0; rt8 < 8; ++rt8) {
    int rt = wave + rt8 * 8;                         // 64 row tiles / 8 waves
    int de0 = q * 1024 + rt * 16;
    v8f acc[4];
#pragma unroll
    for (int j = 0; j < 4; ++j) acc[j] = (v8f){0, 0, 0, 0, 0, 0, 0, 0};
    const float* Arow = Ab + (size_t)(de0 + m) * P0;
#pragma unroll
    for (int k = 0; k < 14; ++k) {
      int p = 4 * k + 2 * h;
      v2f a;
      if (p < P0) a = *(const v2f*)(Arow + p);       // 8B aligned (p even)
      else        a = (v2f){0.f, 0.f};               // pad K 54->56
      const float* vp0 = vs + (size_t)p * NB;
      const float* vp1 = vs + (size_t)(p + 1) * NB;  // rows 54/55 are zeroed
#pragma unroll
      for (int j = 0; j < 4; ++j) {
        v2f bv;
        bv.x = vp0[j * 16 + m];
        bv.y = vp1[j * 16 + m];
        acc[j] = wmma4(a, bv, acc[j]);
      }
    }
#pragma unroll
    for (int j = 0; j < 4; ++j) {
      // C layout: vgpr r -> row de0 + r + 8h ; lane&15 -> batch col j*16+m
      float* out = Tb + (size_t)(j * 16 + m) * 4096 + de0 + 8 * h;
      float4 lo, hi;
      lo.x = acc[j][0]; lo.y = acc[j][1]; lo.z = acc[j][2]; lo.w = acc[j][3];
      hi.x = acc[j][4]; hi.y = acc[j][5]; hi.z = acc[j][6]; hi.w = acc[j][7];
      *(float4*)(out)     = lo;
      *(float4*)(out + 4) = hi;
    }
  }
}

// ---------------------------------------------------------------------------
// K3: chunk products C = T_{7c+1} * ... * T_{7c+7}  (64x64x64 f32 WMMA)
//     grid: (64 batches, 46 chunks). LDS row stride 76 -> conflict-free
//     A (banks 12m+e, halves +-2) and B (rows e/e+2 are 24 banks apart vs a
//     16-wide column span) operand reads; 16B aligned for b128 staging.
// ---------------------------------------------------------------------------
__global__ void __launch_bounds__(256)
k_chunk(const float* __restrict__ T, float* __restrict__ C) {
  const int RS = 76;
  __shared__ float Cb[2][64 * 76];
  __shared__ float Ts[64 * 76];
  int b = blockIdx.x, chunk = blockIdx.y, tid = threadIdx.x;

  const float* T0 = T + ((size_t)(chunk * CHL) * NB + b) * 4096;
  for (int i4 = tid; i4 < 1024; i4 += 256) {
    int r = i4 >> 4, c4 = (i4 & 15) * 4;
    *(float4*)&Cb[0][r * RS + c4] = ((const float4*)T0)[i4];
  }
  int cur = 0;

  int wave = tid >> 5, lane = tid & 31;
  int m = lane & 15, h = lane >> 4;

  for (int s = 1; s < CHL; ++s) {
    const float* Tg = T + ((size_t)(chunk * CHL + s) * NB + b) * 4096;
    if (s + 1 < CHL)
      __builtin_prefetch(T + ((size_t)(chunk * CHL + s + 1) * NB + b) * 4096, 0, 0);
    for (int i4 = tid; i4 < 1024; i4 += 256) {
      int r = i4 >> 4, c4 = (i4 & 15) * 4;
      *(float4*)&Ts[r * RS + c4] = ((const float4*)Tg)[i4];
    }
    __syncthreads();

#pragma unroll
    for (int tt = 0; tt < 2; ++tt) {
      int t = wave * 2 + tt;                         // 16 output tiles / 8 waves
      int ti = t >> 2, tj = t & 3;
      v8f acc = (v8f){0, 0, 0, 0, 0, 0, 0, 0};
#pragma unroll
      for (int k = 0; k < 16; ++k) {
        int e = 4 * k + 2 * h;
        v2f a = *(const v2f*)(&Cb[cur][(ti * 16 + m) * RS + e]);
        v2f bv;
        bv.x = Ts[e * RS + tj * 16 + m];
        bv.y = Ts[(e + 1) * RS + tj * 16 + m];
        acc = wmma4(a, bv, acc);
      }
#pragma unroll
      for (int r = 0; r < 8; ++r)
        Cb[cur ^ 1][(ti * 16 + r + 8 * h) * RS + tj * 16 + m] = acc[r];
    }
    cur ^= 1;
    __syncthreads();
  }

  float* Cout = C + ((size_t)chunk * NB + b) * 4096;
  for (int i4 = tid; i4 < 1024; i4 += 256) {
    int r = i4 >> 4, c4 = (i4 & 15) * 4;
    ((float4*)Cout)[i4] = *(const float4*)&Cb[cur][r * RS + c4];
  }
}

// ---------------------------------------------------------------------------
// K4: per batch: u = C_0 * ... * C_45 * last0 ; h0[o] = sum_p v0[0][p] *
//     sum_d A0_0[o,p,d] u[d].  grid: 64.  Cs stride 65 -> conflict-free.
// ---------------------------------------------------------------------------
__global__ void __launch_bounds__(256)
k_combine(const float* __restrict__ C, const float* __restrict__ vT,
          const float* __restrict__ A0, const float* __restrict__ Alast,
          float* __restrict__ h0) {
  const int CRS = 65;
  __shared__ float u[64], un[64], part[4][64], Cs[64 * 65], v0s[P0];
  int b = blockIdx.x, tid = threadIdx.x;

  if (tid < 64) {
    float s = 0.f;
    for (int p = 0; p < P0; ++p)
      s += Alast[tid * P0 + p] * vT[((size_t)(NSITE0 - 1) * P0P + p) * NB + b];
    u[tid] = s;
  }
  if (tid < P0) v0s[tid] = vT[(size_t)tid * NB + b];
  __syncthreads();

  for (int c = NCHUNK - 1; c >= 0; --c) {
    const float* Cg = C + ((size_t)c * NB + b) * 4096;
    for (int i4 = tid; i4 < 1024; i4 += 256) {
      float4 t = ((const float4*)Cg)[i4];
      int r = i4 >> 4, c4 = (i4 & 15) * 4;
      Cs[r * CRS + c4 + 0] = t.x;
      Cs[r * CRS + c4 + 1] = t.y;
      Cs[r * CRS + c4 + 2] = t.z;
      Cs[r * CRS + c4 + 3] = t.w;
    }
    __syncthreads();
    int d = tid & 63, g = tid >> 6;
    float s = 0.f;
    for (int e = g * 16; e < g * 16 + 16; ++e) s += Cs[d * CRS + e] * u[e];
    part[g][d] = s;
    __syncthreads();
    if (tid < 64) un[tid] = part[0][tid] + part[1][tid] + part[2][tid] + part[3][tid];
    __syncthreads();
    if (tid < 64) u[tid] = un[tid];
    __syncthreads();
  }

  for (int o = tid; o < NSITE0; o += 256) {
    float acc = 0.f;
    const float* Ao = A0 + (size_t)o * P0 * DD;
    for (int p = 0; p < P0; ++p) {
      const float* Ap = Ao + p * DD;
      float w = 0.f;
      for (int d = 0; d < DD; ++d) w += Ap[d] * u[d];
      acc += v0s[p] * w;
    }
    h0[(size_t)b * NSITE0 + o] = acc;
  }
}

// ---------------------------------------------------------------------------
// K5: norm2 (Frobenius norm of layer-2 MPO), single WG of 1024 threads.
// ---------------------------------------------------------------------------
__global__ void __launch_bounds__(1024)
k_norm2(const float* __restrict__ A0, const float* __restrict__ Amid,
        const float* __restrict__ Alast, float* __restrict__ nrm) {
  __shared__ float v[4096], vn[4096], Mp[4096];
  __shared__ float red[1024];
  int tid = threadIdx.x;

  for (int i = tid; i < 4096; i += 1024) {           // v[d,D] from A0_2
    int d = i >> 6, D = i & 63;
    float s = 0.f;
    for (int op = 0; op < 90; ++op) s += A0[op * 64 + d] * A0[op * 64 + D];
    v[i] = s;
  }
  __syncthreads();

  for (int site = 0; site < 2; ++site) {
    for (int i = tid; i < 4096; i += 1024) vn[i] = 0.f;
    __syncthreads();
    const float* A = Amid + (size_t)site * 64 * 64 * 9;
    for (int p = 0; p < 9; ++p) {
      for (int i = tid; i < 4096; i += 1024) {       // Mp[D,e]
        int D = i >> 6, e = i & 63;
        float s = 0.f;
        for (int d = 0; d < 64; ++d) s += v[d * 64 + D] * A[((size_t)d * 64 + e) * 9 + p];
        Mp[i] = s;
      }
      __syncthreads();
      for (int i = tid; i < 4096; i += 1024) {       // vn[e,E] += ...
        int e = i >> 6, E = i & 63;
        float s = 0.f;
        for (int D = 0; D < 64; ++D) s += Mp[D * 64 + e] * A[((size_t)D * 64 + E) * 9 + p];
        vn[i] += s;
      }
      __syncthreads();
    }
    for (int i = tid; i < 4096; i += 1024) v[i] = vn[i];
    __syncthreads();
  }

  float s = 0.f;
  for (int i = tid; i < 4096; i += 1024) {
    int d = i >> 6, D = i & 63;
    float t = 0.f;
    for (int p = 0; p < 9; ++p) t += Alast[d * 9 + p] * Alast[D * 9 + p];
    s += v[i] * t;
  }
  red[tid] = s;
  __syncthreads();
  for (int k = 512; k > 0; k >>= 1) {
    if (tid < k) red[tid] += red[tid + k];
    __syncthreads();
  }
  if (tid == 0) nrm[0] = sqrtf(red[0]);
}

// ---------------------------------------------------------------------------
// K6: layers 1 and 2 (tiny), per-batch. grid: 64
// ---------------------------------------------------------------------------
__global__ void __launch_bounds__(256)
k_tail(const float* __restrict__ h0g,
       const float* __restrict__ A0_1, const float* __restrict__ Amid_1,
       const float* __restrict__ Alast_1,
       const float* __restrict__ A0_2, const float* __restrict__ Amid_2,
       const float* __restrict__ Alast_2,
       const float* __restrict__ nrmp, float* __restrict__ out) {
  __shared__ float h0[NSITE0], v1[36 * 9], u[64], un[64], part[4][64];
  __shared__ float h1[36], v2[4 * 9];
  int b = blockIdx.x, tid = threadIdx.x;

  for (int i = tid; i < NSITE0; i += 256) h0[i] = h0g[(size_t)b * NSITE0 + i];
  __syncthreads();

  if (tid < 36) {                                    // squeeze 18x18 -> 36 x 9
    int i = tid / 6, j = tid % 6;
    float nrm2 = 0.f, vals[9];
    for (int a = 0; a < 3; ++a)
      for (int c = 0; c < 3; ++c) {
        float xv = h0[(3 * i + a) * 18 + 3 * j + c];
        vals[a * 3 + c] = xv;
        nrm2 += xv * xv;
      }
    float inv = rsqrtf(nrm2);
    for (int k = 0; k < 9; ++k) v1[tid * 9 + k] = vals[k] * inv;
  }
  __syncthreads();

  if (tid < 64) {                                    // u = last1
    float s = 0.f;
    for (int p = 0; p < 9; ++p) s += Alast_1[tid * 9 + p] * v1[35 * 9 + p];
    u[tid] = s;
  }
  __syncthreads();

  for (int n = 33; n >= 0; --n) {                    // u = T_n u (layer 1)
    int d = tid & 63, g = tid >> 6;
    const float* A = Amid_1 + ((size_t)n * 64 + d) * 64 * 9;
    const float* vp = v1 + (n + 1) * 9;
    float s = 0.f;
    for (int e = g * 16; e < g * 16 + 16; ++e) {
      const float* Ap = A + e * 9;
      float t = 0.f;
      for (int p = 0; p < 9; ++p) t += Ap[p] * vp[p];
      s += u[e] * t;
    }
    part[g][d] = s;
    __syncthreads();
    if (tid < 64) un[tid] = part[0][tid] + part[1][tid] + part[2][tid] + part[3][tid];
    __syncthreads();
    if (tid < 64) u[tid] = un[tid];
    __syncthreads();
  }

  if (tid < 36) {                                    // h1
    float acc = 0.f;
    for (int p = 0; p < 9; ++p) {
      const float* Ap = A0_1 + ((size_t)tid * 9 + p) * 64;
      float w = 0.f;
      for (int d = 0; d < 64; ++d) w += Ap[d] * u[d];
      acc += v1[p] * w;
    }
    h1[tid] = acc;
  }
  __syncthreads();

  if (tid < 4) {                                     // squeeze 6x6 -> 4 x 9
    int i = tid >> 1, j = tid & 1;
    float nrm2 = 0.f, vals[9];
    for (int a = 0; a < 3; ++a)
      for (int c = 0; c < 3; ++c) {
        float xv = h1[(3 * i + a) * 6 + 3 * j + c];
        vals[a * 3 + c] = xv;
        nrm2 += xv * xv;
      }
    float inv = rsqrtf(nrm2);
    for (int k = 0; k < 9; ++k) v2[tid * 9 + k] = vals[k] * inv;
  }
  __syncthreads();

  if (tid < 64) {                                    // u = last2
    float s = 0.f;
    for (int p = 0; p < 9; ++p) s += Alast_2[tid * 9 + p] * v2[3 * 9 + p];
    u[tid] = s;
  }
  __syncthreads();

  for (int n = 1; n >= 0; --n) {                     // u = T_n u (layer 2)
    int d = tid & 63, g = tid >> 6;
    const float* A = Amid_2 + ((size_t)n * 64 + d) * 64 * 9;
    const float* vp = v2 + (n + 1) * 9;
    float s = 0.f;
    for (int e = g * 16; e < g * 16 + 16; ++e) {
      const float* Ap = A + e * 9;
      float t = 0.f;
      for (int p = 0; p < 9; ++p) t += Ap[p] * vp[p];
      s += u[e] * t;
    }
    part[g][d] = s;
    __syncthreads();
    if (tid < 64) un[tid] = part[0][tid] + part[1][tid] + part[2][tid] + part[3][tid];
    __syncthreads();
    if (tid < 64) u[tid] = un[tid];
    __syncthreads();
  }

  if (tid < 10) {
    float acc = 0.f;
    for (int p = 0; p < 9; ++p) {
      const float* Ap = A0_2 + ((size_t)tid * 9 + p) * 64;
      float w = 0.f;
      for (int d = 0; d < 64; ++d) w += Ap[d] * u[d];
      acc += v2[p] * w;
    }
    out[b * 10 + tid] = acc / nrmp[0];
  }
}

// ---------------------------------------------------------------------------
extern "C" void kernel_launch(void* const* d_in, const int* in_sizes, int n_in,
                              void* d_out, int out_size, void* d_ws, size_t ws_size,
                              hipStream_t stream) {
  (void)in_sizes; (void)n_in; (void)out_size; (void)ws_size;
  const float* x       = (const float*)d_in[0];
  const float* A0_0    = (const float*)d_in[1];
  const float* Amid_0  = (const float*)d_in[2];
  const float* Alast_0 = (const float*)d_in[3];
  const float* A0_1    = (const float*)d_in[4];
  const float* Amid_1  = (const float*)d_in[5];
  const float* Alast_1 = (const float*)d_in[6];
  const float* A0_2    = (const float*)d_in[7];
  const float* Amid_2  = (const float*)d_in[8];
  const float* Alast_2 = (const float*)d_in[9];

  float* ws = (float*)d_ws;
  float* vT = ws + OFF_VT;
  float* T  = ws + OFF_T;
  float* C  = ws + OFF_C;
  float* h0 = ws + OFF_H0;
  float* nm = ws + OFF_NRM;

  k_embed<<<(NB * NSITE0 * 27 + 255) / 256, 256, 0, stream>>>(x, vT);
  k_tmat<<<dim3(4, NMID0), 256, 0, stream>>>(Amid_0, vT, T);
  k_chunk<<<dim3(NB, NCHUNK), 256, 0, stream>>>(T, C);
  k_combine<<<NB, 256, 0, stream>>>(C, vT, A0_0, Alast_0, h0);
  k_norm2<<<1, 1024, 0, stream>>>(A0_2, Amid_2, Alast_2, nm);
  k_tail<<<NB, 256, 0, stream>>>(h0, A0_1, Amid_1, Alast_1,
                                 A0_2, Amid_2, Alast_2, nm, (float*)d_out);
}